// QuantumStateEncoder_1030792151630
// MI455X (gfx1250) — compile-verified
//
#include <hip/hip_runtime.h>
#include <math.h>

typedef __attribute__((ext_vector_type(2))) float v2f;
typedef __attribute__((ext_vector_type(8))) float v8f;

#define NQ 10          // qubits
#define NQP 16         // padded qubit dim
#define DIN 512
#define DH 64
#define DOUT 256
#define BATCH 4096

__device__ __forceinline__ float shflx(float v, int mask) {
  return __shfl_xor(v, mask, 32);
}

// ---------------------------------------------------------------------------
// Prep: build transposed / zero-padded weight copies so the WMMA loops are
// completely branch-free with contiguous per-lane (K,K+1) pairs.
//   WinT [16][512] : WinT[n][k] = (n<10) ? W_in[k][n] : 0
//   W1T  [64][16]  : W1T[n][k]  = (k<10) ? W1[k][n]  : 0
//   W2T  [256][64] : W2T[n][k]  = W2[k][n]
// ---------------------------------------------------------------------------
__global__ __launch_bounds__(256)
void k_prep(const float* __restrict__ Win, const float* __restrict__ W1,
            const float* __restrict__ W2, float* __restrict__ WinT,
            float* __restrict__ W1T, float* __restrict__ W2T) {
  const int t = blockIdx.x * 256 + threadIdx.x;
  if (t < NQP * DIN) {
    const int n = t >> 9, k = t & 511;
    WinT[t] = (n < NQ) ? Win[k * NQ + n] : 0.0f;
  }
  const int t1 = t - NQP * DIN;
  if (t1 >= 0 && t1 < DH * NQP) {
    const int n = t1 >> 4, k = t1 & 15;
    W1T[t1] = (k < NQ) ? W1[k * DH + n] : 0.0f;
  }
  const int t2 = t1 - DH * NQP;
  if (t2 >= 0 && t2 < DOUT * DH) {
    const int n = t2 >> 6, k = t2 & 63;
    W2T[t2] = W2[k * DOUT + n];
  }
}

// ---------------------------------------------------------------------------
// Kernel A: angles = x @ W_in + b_in   ([4096,512] x [512,10] -> [4096,10])
// 4 waves / block, one 16-row tile per wave. WMMA f32 16x16x4.
// ---------------------------------------------------------------------------
__global__ __launch_bounds__(128)
void k_angles(const float* __restrict__ x, const float* __restrict__ WinT,
              const float* __restrict__ bin, float* __restrict__ angles) {
  const int lane = threadIdx.x & 31;
  const int wave = threadIdx.x >> 5;
  const int m0 = blockIdx.x * 64 + wave * 16;
  const int l = lane & 15;           // A row / B column within tile
  const int koff = (lane >> 4) * 2;  // K sub-offset for upper half-wave

  const float* arow = x + (size_t)(m0 + l) * DIN;
  const float* bcol = WinT + (size_t)l * DIN;

  v8f acc = {};
  for (int kb = 0; kb < DIN; kb += 64) {
    __builtin_prefetch(arow + kb + 64, 0, 1);  // global_prefetch_b8 (speculative)
#pragma unroll
    for (int k0 = kb; k0 < kb + 64; k0 += 4) {
      const int k = k0 + koff;
      v2f a = *(const v2f*)(arow + k);
      v2f b = *(const v2f*)(bcol + k);
      acc = __builtin_amdgcn_wmma_f32_16x16x4_f32(false, a, false, b,
                                                  (short)0, acc, false, false);
    }
  }
  // C/D layout: VGPR r, lanes 0-15 -> M=r, N=lane; lanes 16-31 -> M=8+r
  const int rbase = (lane >> 4) * 8;
  if (l < NQ) {
#pragma unroll
    for (int r = 0; r < 8; ++r) {
      angles[(m0 + rbase + r) * NQ + l] = acc[r] + bin[l];
    }
  }
}

// ---------------------------------------------------------------------------
// Kernel B: 10-qubit statevector simulation, one batch element per wave.
// State: 1024 complex amps; index bits[9:5]=lane, bits[4:0]=register slot.
// Wire w acts on bit position (9-w): pos>=5 -> cross-lane shfl_xor butterfly,
// pos<5 -> in-register pair butterfly (fully unrolled, constant indices).
// Writes z padded to 16 columns (cols 10..15 = 0) for branch-free GEMM2.
// ---------------------------------------------------------------------------
__global__ __launch_bounds__(128)
void k_quantum(const float* __restrict__ angles, const float* __restrict__ qw,
               float* __restrict__ z) {
  const int lane = threadIdx.x & 31;
  const int wave = threadIdx.x >> 5;
  const int bidx = blockIdx.x * 4 + wave;

  float sr[32], si[32];
#pragma unroll
  for (int j = 0; j < 32; ++j) { sr[j] = 0.0f; si[j] = 0.0f; }
  if (lane == 0) sr[0] = 1.0f;   // |0...0>

  // ---- RX encoding layer: gate [[c, -i s],[-i s, c]] on wire i ----
#pragma unroll
  for (int i = 0; i < NQ; ++i) {
    float s, c;
    __sincosf(0.5f * angles[bidx * NQ + i], &s, &c);
    const int pos = 9 - i;
    if (pos >= 5) {
      const int msk = 1 << (pos - 5);
#pragma unroll
      for (int j = 0; j < 32; ++j) {
        float pr = shflx(sr[j], msk);
        float pi = shflx(si[j], msk);
        float nr = c * sr[j] + s * pi;
        float ni = c * si[j] - s * pr;
        sr[j] = nr; si[j] = ni;
      }
    } else {
      const int bit = 1 << pos;
#pragma unroll
      for (int j = 0; j < 32; ++j) {
        if (!(j & bit)) {
          const int k = j | bit;
          float a0r = sr[j], a0i = si[j], a1r = sr[k], a1i = si[k];
          sr[j] = c * a0r + s * a1i;
          si[j] = c * a0i - s * a1r;
          sr[k] = c * a1r + s * a0i;
          si[k] = c * a1i - s * a0r;
        }
      }
    }
  }

  // ---- entangling layers: Rot(phi,theta,omega) per wire + CNOT ring ----
#pragma unroll
  for (int lyr = 0; lyr < 2; ++lyr) {
#pragma unroll
    for (int i = 0; i < NQ; ++i) {
      const float phi = qw[(lyr * NQ + i) * 3 + 0];
      const float th  = qw[(lyr * NQ + i) * 3 + 1];
      const float om  = qw[(lyr * NQ + i) * 3 + 2];
      float st, ct, sp, cp, sm, cm;
      __sincosf(0.5f * th, &st, &ct);
      __sincosf(0.5f * (phi + om), &sp, &cp);   // epo = (cp, -sp)
      __sincosf(0.5f * (phi - om), &sm, &cm);   // epm = (cm, -sm)
      const float g00r =  cp * ct, g00i = -sp * ct;  // epo*ct
      const float g01r = -cm * st, g01i = -sm * st;  // -conj(epm)*st
      const float g10r =  cm * st, g10i = -sm * st;  // epm*st
      const float g11r =  cp * ct, g11i =  sp * ct;  // conj(epo)*ct
      const int pos = 9 - i;
      if (pos >= 5) {
        const int msk = 1 << (pos - 5);
        const bool hi = (lane >> (pos - 5)) & 1;
        const float ar = hi ? g11r : g00r, ai = hi ? g11i : g00i; // * self
        const float br = hi ? g10r : g01r, bi = hi ? g10i : g01i; // * partner
#pragma unroll
        for (int j = 0; j < 32; ++j) {
          float pr = shflx(sr[j], msk);
          float pi = shflx(si[j], msk);
          float nr = ar * sr[j] - ai * si[j] + br * pr - bi * pi;
          float ni = ar * si[j] + ai * sr[j] + br * pi + bi * pr;
          sr[j] = nr; si[j] = ni;
        }
      } else {
        const int bit = 1 << pos;
#pragma unroll
        for (int j = 0; j < 32; ++j) {
          if (!(j & bit)) {
            const int k = j | bit;
            float a0r = sr[j], a0i = si[j], a1r = sr[k], a1i = si[k];
            sr[j] = g00r * a0r - g00i * a0i + g01r * a1r - g01i * a1i;
            si[j] = g00r * a0i + g00i * a0r + g01r * a1i + g01i * a1r;
            sr[k] = g10r * a0r - g10i * a0i + g11r * a1r - g11i * a1i;
            si[k] = g10r * a0i + g10i * a0r + g11r * a1i + g11i * a1r;
          }
        }
      }
    }
    // CNOT ring, range r = lyr % 9 + 1
    const int rr = lyr + 1;
#pragma unroll
    for (int i = 0; i < NQ; ++i) {
      const int pc = 9 - i;                 // control bit position
      const int pt = 9 - ((i + rr) % NQ);   // target bit position
      if (pc >= 5 && pt >= 5) {
        const int msk = 1 << (pt - 5);
        const bool ctrl = (lane >> (pc - 5)) & 1;
#pragma unroll
        for (int j = 0; j < 32; ++j) {
          float pr = shflx(sr[j], msk);
          float pi = shflx(si[j], msk);
          sr[j] = ctrl ? pr : sr[j];
          si[j] = ctrl ? pi : si[j];
        }
      } else if (pc >= 5) {  // control=lane bit, target=register bit
        const bool ctrl = (lane >> (pc - 5)) & 1;
        const int bit = 1 << pt;
#pragma unroll
        for (int j = 0; j < 32; ++j) {
          if (!(j & bit)) {
            const int k = j | bit;
            float ar = sr[j], ai = si[j], br = sr[k], bi = si[k];
            sr[j] = ctrl ? br : ar; si[j] = ctrl ? bi : ai;
            sr[k] = ctrl ? ar : br; si[k] = ctrl ? ai : bi;
          }
        }
      } else if (pt >= 5) {  // control=register bit, target=lane bit
        const int msk = 1 << (pt - 5);
        const int bit = 1 << pc;
#pragma unroll
        for (int j = 0; j < 32; ++j) {
          if (j & bit) {
            sr[j] = shflx(sr[j], msk);
            si[j] = shflx(si[j], msk);
          }
        }
      } else {               // both register bits: pure register permute
        const int cb = 1 << pc, tb = 1 << pt;
#pragma unroll
        for (int j = 0; j < 32; ++j) {
          if ((j & cb) && !(j & tb)) {
            const int k = j | tb;
            float tr = sr[j], ti = si[j];
            sr[j] = sr[k]; si[j] = si[k];
            sr[k] = tr;    si[k] = ti;
          }
        }
      }
    }
  }

  // ---- Z expectations: z[q] = sum_a |amp_a|^2 * (1 - 2*bit_{9-q}(a)) ----
  float p[32];
#pragma unroll
  for (int j = 0; j < 32; ++j) p[j] = sr[j] * sr[j] + si[j] * si[j];

  float ptot = 0.0f;
#pragma unroll
  for (int j = 0; j < 32; ++j) ptot += p[j];

  float part[NQ];
#pragma unroll
  for (int q = 0; q < 5; ++q) {          // sign lives on a lane bit
    const int L = 4 - q;
    part[q] = ((lane >> L) & 1) ? -ptot : ptot;
  }
#pragma unroll
  for (int q = 5; q < NQ; ++q) {         // sign lives on a register bit
    const int bit = 1 << (9 - q);
    float acc = 0.0f;
#pragma unroll
    for (int j = 0; j < 32; ++j) acc += (j & bit) ? -p[j] : p[j];
    part[q] = acc;
  }
#pragma unroll
  for (int q = 0; q < NQ; ++q) {
    float v = part[q];
#pragma unroll
    for (int off = 16; off >= 1; off >>= 1) v += shflx(v, off);
    if (lane == 0) z[bidx * NQP + q] = v;
  }
  if (lane < NQP - NQ) z[bidx * NQP + NQ + lane] = 0.0f;  // pad cols
}

// ---------------------------------------------------------------------------
// Kernel C: out = tanh(relu(z@W1+b1)@W2+b2). 4 waves / block, one 16-row
// tile per wave. W2T (64KB) is staged once per workgroup into LDS with
// GLOBAL_LOAD_ASYNC_TO_LDS_B128 (weights-stationary, shared by all 4 waves),
// overlapped with GEMM2; GEMM3 reads B fragments via ds_load_b64.
// ---------------------------------------------------------------------------
__global__ __launch_bounds__(128)
void k_mlp(const float* __restrict__ z, const float* __restrict__ W1T,
           const float* __restrict__ b1, const float* __restrict__ W2T,
           const float* __restrict__ b2, float* __restrict__ out) {
  __shared__ float w2s[DOUT * DH];   // 64 KB, W2 transposed copy
  __shared__ float hs[4][16 * DH];   // 16 KB, per-wave h tiles
  const int lane = threadIdx.x & 31;
  const int wave = threadIdx.x >> 5;
  const int m0 = blockIdx.x * 64 + wave * 16;
  const int l = lane & 15;
  const int koff = (lane >> 4) * 2;
  const int rbase = (lane >> 4) * 8;
  float* hw = hs[wave];

  // ---- kick off async W2T -> LDS (128 lanes x 16B x 32 rounds = 64KB) ----
  {
    const unsigned lbase = (unsigned)(uintptr_t)(void*)w2s;  // LDS byte offset
    const unsigned long long gbase = (unsigned long long)(uintptr_t)W2T;
#pragma unroll
    for (int it = 0; it < 32; ++it) {
      const unsigned idx = (unsigned)(it * 128 + threadIdx.x);
      const unsigned la = lbase + idx * 16u;
      const unsigned long long ga = gbase + (unsigned long long)idx * 16ull;
      asm volatile("global_load_async_to_lds_b128 %0, %1, off"
                   :: "v"(la), "v"(ga) : "memory");
    }
  }

  // ---- GEMM2 (overlaps the async fill): [16,16] x [16,64], K padded ----
  const float* arow2 = z + (size_t)(m0 + l) * NQP;
#pragma unroll
  for (int nt = 0; nt < 4; ++nt) {
    const float* bcol2 = W1T + (size_t)(nt * 16 + l) * NQP;
    v8f acc = {};
#pragma unroll
    for (int k0 = 0; k0 < NQP; k0 += 4) {
      const int k = k0 + koff;
      v2f a = *(const v2f*)(arow2 + k);
      v2f b = *(const v2f*)(bcol2 + k);
      acc = __builtin_amdgcn_wmma_f32_16x16x4_f32(false, a, false, b,
                                                  (short)0, acc, false, false);
    }
    const int col = nt * 16 + l;
#pragma unroll
    for (int r = 0; r < 8; ++r) {
      float v = acc[r] + b1[col];
      hw[(rbase + r) * DH + col] = v > 0.0f ? v : 0.0f;
    }
  }

  // ---- wait for the async fill, then barrier so all waves see w2s ----
  asm volatile("s_wait_asynccnt 0x0" ::: "memory");
  __syncthreads();

  // ---- GEMM3: [16,64] x [64,256], B from LDS ----
  const float* arow3 = hw + l * DH;
#pragma unroll
  for (int nt = 0; nt < 16; ++nt) {
    const float* bcol3 = w2s + (nt * 16 + l) * DH;
    v8f acc = {};
#pragma unroll
    for (int k0 = 0; k0 < DH; k0 += 4) {
      const int k = k0 + koff;
      v2f a = *(const v2f*)(arow3 + k);
      v2f b = *(const v2f*)(bcol3 + k);
      acc = __builtin_amdgcn_wmma_f32_16x16x4_f32(false, a, false, b,
                                                  (short)0, acc, false, false);
    }
    const int col = nt * 16 + l;
#pragma unroll
    for (int r = 0; r < 8; ++r) {
      out[(m0 + rbase + r) * DOUT + col] = tanhf(acc[r] + b2[col]);
    }
  }
}

// ---------------------------------------------------------------------------
extern "C" void kernel_launch(void* const* d_in, const int* in_sizes, int n_in,
                              void* d_out, int out_size, void* d_ws, size_t ws_size,
                              hipStream_t stream) {
  const float* x   = (const float*)d_in[0];
  const float* Win = (const float*)d_in[1];
  const float* bin = (const float*)d_in[2];
  const float* qw  = (const float*)d_in[3];
  const float* W1  = (const float*)d_in[4];
  const float* b1  = (const float*)d_in[5];
  const float* W2  = (const float*)d_in[6];
  const float* b2  = (const float*)d_in[7];
  float* out = (float*)d_out;

  float* angles = (float*)d_ws;                   // [4096,10]
  float* zpad   = angles + (size_t)BATCH * NQ;    // [4096,16]
  float* WinT   = zpad + (size_t)BATCH * NQP;     // [16,512]
  float* W1T    = WinT + (size_t)NQP * DIN;       // [64,16]
  float* W2T    = W1T + (size_t)DH * NQP;         // [256,64]

  k_prep   <<<100, 256, 0, stream>>>(Win, W1, W2, WinT, W1T, W2T);
  k_angles <<<BATCH / 64, 128, 0, stream>>>(x, WinT, bin, angles);
  k_quantum<<<BATCH / 4, 128, 0, stream>>>(angles, qw, zpad);
  k_mlp    <<<BATCH / 64, 128, 0, stream>>>(zpad, W1T, b1, W2T, b2, out);
}